// SinkhornKnopp_18889266167813
// MI455X (gfx1250) — compile-verified
//
#include <hip/hip_runtime.h>
#include <math.h>
#include <stdint.h>

// Sinkhorn-Knopp, B=16384 x K=4096 fp32, diagonal-scaling formulation.
#define B_DIM 16384
#define K_DIM 4096
#define NSTRIP 64
#define STRIP_ROWS (B_DIM / NSTRIP) // 256 rows per column-reduction block
#define ROWS 16                     // rows per WMMA row-sum block
#define CHUNK 512                   // k-elements staged in LDS per chunk
#define LDS_STRIDE (CHUNK + 4)      // pad: stride%64==4 -> conflict-free ds_read
#define SCALE2 28.853900817779268f  // log2(e) / 0.05
#define R_MARG (1.0f / 4096.0f)
#define C_MARG (1.0f / 16384.0f)

typedef __attribute__((ext_vector_type(2))) float v2f;
typedef __attribute__((ext_vector_type(8))) float v8f;

// Raw v_exp_f32: args are <= 0 here; sub-(-126) inputs flush to 0 which is
// numerically irrelevant against the O(1) sums. Avoids the 7-op range fixup.
__device__ __forceinline__ float fast_exp2(float x) {
  return __builtin_amdgcn_exp2f(x);
}

__device__ __forceinline__ unsigned f32_key(float f) {
  unsigned u = __float_as_uint(f);
  return (u & 0x80000000u) ? ~u : (u | 0x80000000u); // monotone order-preserving map
}
__device__ __forceinline__ float key_f32(unsigned k) {
  unsigned u = (k & 0x80000000u) ? (k & 0x7fffffffu) : ~k;
  return __uint_as_float(u);
}

__global__ __launch_bounds__(256) void sk_init(unsigned* maxKey, float* colg) {
  int i = blockIdx.x * 256 + threadIdx.x;
  if (i < B_DIM) colg[i] = 1.0f;   // gamma starts at 1 (global norm cancels)
  if (i == 0) *maxKey = 0u;        // key 0 == most-negative float
}

__global__ __launch_bounds__(256) void sk_max(const float* __restrict__ x,
                                              unsigned* __restrict__ maxKey) {
  __shared__ float red[256];
  float m = -3.402823466e38f;
  const size_t n = (size_t)B_DIM * K_DIM;
  const size_t stride = (size_t)gridDim.x * 256u;
  for (size_t i = (size_t)blockIdx.x * 256u + threadIdx.x; i < n; i += stride)
    m = fmaxf(m, x[i]);
  red[threadIdx.x] = m;
  __syncthreads();
  for (int off = 128; off > 0; off >>= 1) {
    if (threadIdx.x < off)
      red[threadIdx.x] = fmaxf(red[threadIdx.x], red[threadIdx.x + off]);
    __syncthreads();
  }
  if (threadIdx.x == 0) atomicMax(maxKey, f32_key(red[0])); // max is exact: deterministic
}

__global__ void sk_finish_max(const unsigned* __restrict__ maxKey,
                              float* __restrict__ bias2) {
  *bias2 = -key_f32(*maxKey) * SCALE2; // exp((x-max)/eps) == exp2(x*SCALE2 + bias2)
}

// partial[strip][k] = sum_{b in strip} exp((x[b,k]-max)/eps) * gamma[b]
__global__ __launch_bounds__(256) void sk_col_reduce(const float* __restrict__ logits,
                                                     const float* __restrict__ bias2p,
                                                     const float* __restrict__ gamma,
                                                     float* __restrict__ partial) {
  __shared__ float g[STRIP_ROWS];
  const int k = blockIdx.x * 256 + threadIdx.x;
  const int b0 = blockIdx.y * STRIP_ROWS;
  const float bias2 = bias2p[0];
  g[threadIdx.x] = gamma[b0 + threadIdx.x]; // STRIP_ROWS == blockDim.x
  __syncthreads();
  float acc = 0.f;
  const float* p = logits + (size_t)b0 * K_DIM + k;
#pragma unroll 4
  for (int b = 0; b < STRIP_ROWS; ++b) {
    float e = fast_exp2(__builtin_fmaf(p[(size_t)b * K_DIM], SCALE2, bias2));
    acc = __builtin_fmaf(g[b], e, acc);
  }
  partial[(size_t)blockIdx.y * K_DIM + k] = acc; // direct store: deterministic
}

__global__ __launch_bounds__(256) void sk_update_rho(const float* __restrict__ partial,
                                                     float* __restrict__ rho) {
  const int k = blockIdx.x * 256 + threadIdx.x;
  float s = 0.f;
  for (int j = 0; j < NSTRIP; ++j) s += partial[(size_t)j * K_DIM + k];
  rho[k] = R_MARG / s; // rho_new = r / (Q0 @ gamma), old rho cancels exactly
}

// Per-16-row weighted sums via V_WMMA_F32_16X16X4_F32: D = A x B + C where
// A[m,j] = exp((x[b0+m, k0+j]-max)/eps) and B[j,n] = rho[k0+j] (uniform in n).
// Raw logits tiles are DMA'd global->LDS with GLOBAL_LOAD_ASYNC_TO_LDS_B128
// (ASYNCcnt), bypassing VGPRs; exp is applied on the 2 A-values/lane at
// fragment-read time. A fragment (ISA 7.12.2, 32-bit 16x4): lanes 0-15 ->
// M=lane, K={k,k+1}; lanes 16-31 -> M=lane-16, K={k+2,k+3}. All columns of D
// equal the row sums; C/D layout: VGPR r holds M=r (lanes 0-15), M=r+8 (16-31).
__device__ __forceinline__ void sk_row_sums_wmma(const float* __restrict__ logits,
                                                 const float* __restrict__ rho,
                                                 float bias2, int b0,
                                                 float tile[][LDS_STRIDE],
                                                 float* ldsRho,
                                                 float wsum[][16],
                                                 float* srow) {
  const int tid = threadIdx.x;
  const int wave = tid >> 5;
  const int lane = tid & 31;
  v8f acc = {0.f, 0.f, 0.f, 0.f, 0.f, 0.f, 0.f, 0.f};
  const unsigned ldsTile = (unsigned)(uintptr_t)&tile[0][0]; // low 32b == LDS offset
  for (int kc = 0; kc < K_DIM; kc += CHUNK) {
    // Async DMA: 16 rows x CHUNK raw floats -> LDS, 16B per lane per issue.
    // Both global and LDS addresses are 16B aligned (k multiple of 4,
    // row stride 516*4 = 16*129 bytes).
#pragma unroll
    for (int r = 0; r < (ROWS * CHUNK) / (256 * 4); ++r) { // 8 issues/thread
      const int e = (tid + r * 256) * 4;  // first element of this 16B packet
      const int m = e >> 9;               // / CHUNK
      const int k = e & (CHUNK - 1);
      const uint64_t ga =
          (uint64_t)(uintptr_t)(logits + (size_t)(b0 + m) * K_DIM + kc + k);
      const unsigned la = ldsTile + (unsigned)(m * LDS_STRIDE + k) * 4u;
      asm volatile("global_load_async_to_lds_b128 %0, %1, off"
                   :: "v"(la), "v"(ga) : "memory");
    }
    // Stage rho chunk for the B operand (tiny, regular ds_store).
    ldsRho[tid] = rho[kc + tid];
    ldsRho[tid + 256] = rho[kc + tid + 256];
    asm volatile("s_wait_asynccnt 0" ::: "memory");
    __syncthreads();

    const int m = lane & 15;
    const int kw = wave * (CHUNK / 8);     // 64 k per wave
    const int koff = (lane >> 4) << 1;     // 0 or 2
    const float* srcA = &tile[m][kw + koff];
    const float* srcB = &ldsRho[kw + koff];
#pragma unroll 4
    for (int k4 = 0; k4 < CHUNK / 8; k4 += 4) {
      v2f xa = *(const v2f*)(srcA + k4);   // ds_read_b64, raw logits
      v2f rb = *(const v2f*)(srcB + k4);   // ds_read_b64, rho broadcast
      v2f a;
      a.x = fast_exp2(__builtin_fmaf(xa.x, SCALE2, bias2));
      a.y = fast_exp2(__builtin_fmaf(xa.y, SCALE2, bias2));
      acc = __builtin_amdgcn_wmma_f32_16x16x4_f32(
          /*neg_a=*/false, a, /*neg_b=*/false, rb,
          /*c_mod=*/(short)0, acc, /*reuse_a=*/false, /*reuse_b=*/false);
    }
    __syncthreads();
  }
  if (lane == 0) {
#pragma unroll
    for (int r = 0; r < 8; ++r) wsum[wave][r] = acc[r];      // rows 0..7
  }
  if (lane == 16) {
#pragma unroll
    for (int r = 0; r < 8; ++r) wsum[wave][8 + r] = acc[r];  // rows 8..15
  }
  __syncthreads();
  if (tid < 16) {
    float s = 0.f;
#pragma unroll
    for (int w = 0; w < 8; ++w) s += wsum[w][tid];
    srow[tid] = s;
  }
  __syncthreads();
}

__global__ __launch_bounds__(256) void sk_col_sums(const float* __restrict__ logits,
                                                   const float* __restrict__ bias2p,
                                                   const float* __restrict__ rho,
                                                   float* __restrict__ colAcc) {
  __shared__ __align__(16) float tile[ROWS][LDS_STRIDE];
  __shared__ __align__(16) float ldsRho[CHUNK + 8];
  __shared__ float wsum[8][16];
  __shared__ float srow[16];
  const int b0 = blockIdx.x * ROWS;
  sk_row_sums_wmma(logits, rho, bias2p[0], b0, tile, ldsRho, wsum, srow);
  if (threadIdx.x < 16) colAcc[b0 + threadIdx.x] = srow[threadIdx.x];
}

__global__ __launch_bounds__(256) void sk_update_gamma(float* __restrict__ colg) {
  const int b = blockIdx.x * 256 + threadIdx.x;
  colg[b] = C_MARG / colg[b]; // gamma_new = c / (rho @ Q0), old gamma cancels
}

__global__ __launch_bounds__(256) void sk_final(const float* __restrict__ logits,
                                                const float* __restrict__ bias2p,
                                                const float* __restrict__ rho,
                                                float* __restrict__ out) {
  __shared__ __align__(16) float tile[ROWS][LDS_STRIDE];
  __shared__ __align__(16) float ldsRho[CHUNK + 8];
  __shared__ float wsum[8][16];
  __shared__ float srow[16];
  const int b0 = blockIdx.x * ROWS;
  const float bias2 = bias2p[0];
  sk_row_sums_wmma(logits, rho, bias2, b0, tile, ldsRho, wsum, srow);
  // out[b,k] = rho[k]*Q0[k,b] / sum_k rho*Q0  (gamma cancels in the reference)
  for (int i = threadIdx.x; i < ROWS * K_DIM; i += 256) {
    const int m = i >> 12;              // / K_DIM
    const int k = i & (K_DIM - 1);
    const float x = logits[(size_t)(b0 + m) * K_DIM + k];
    const float v = rho[k] * fast_exp2(__builtin_fmaf(x, SCALE2, bias2));
    out[(size_t)(b0 + m) * K_DIM + k] = v / srow[m];
  }
}

extern "C" void kernel_launch(void* const* d_in, const int* in_sizes, int n_in,
                              void* d_out, int out_size, void* d_ws, size_t ws_size,
                              hipStream_t stream) {
  const float* logits = (const float*)d_in[0];
  float* out = (float*)d_out;

  // Workspace layout (floats): [0]=maxKey(uint) [1]=bias2 [8..) partial, rho, colg
  float* wsf = (float*)d_ws;
  unsigned* maxKey = (unsigned*)wsf;
  float* bias2 = wsf + 1;
  float* partial = wsf + 8;                                // NSTRIP*K floats
  float* rho = partial + (size_t)NSTRIP * K_DIM;           // K floats
  float* colg = rho + K_DIM;                               // B floats

  sk_init<<<B_DIM / 256, 256, 0, stream>>>(maxKey, colg);
  sk_max<<<2048, 256, 0, stream>>>(logits, maxKey);
  sk_finish_max<<<1, 1, 0, stream>>>(maxKey, bias2);

  const dim3 gA(K_DIM / 256, NSTRIP);
  for (int it = 0; it < 3; ++it) {
    sk_col_reduce<<<gA, 256, 0, stream>>>(logits, bias2, colg, partial);
    sk_update_rho<<<K_DIM / 256, 256, 0, stream>>>(partial, rho);
    if (it < 2) { // iteration 3's gamma cancels in the final column normalization
      sk_col_sums<<<B_DIM / ROWS, 256, 0, stream>>>(logits, bias2, rho, colg);
      sk_update_gamma<<<B_DIM / 256, 256, 0, stream>>>(colg);
    }
  }
  sk_final<<<B_DIM / ROWS, 256, 0, stream>>>(logits, bias2, rho, out);
}